// AttentionFixModel_88012469830352
// MI455X (gfx1250) — compile-verified
//
#include <hip/hip_runtime.h>

typedef __attribute__((ext_vector_type(2))) float v2f;
typedef __attribute__((ext_vector_type(8))) float v8f;

static constexpr int kBT   = 256;   // B*T rows
static constexpr int kD    = 384;   // feature dim == QDIM
static constexpr int kNP   = 256;   // patches per (b,t)
static constexpr int kTopK = 16;

// ---------------------------------------------------------------------------
// GEMM: out[r,c] = sum_k A[r,k] * W[k,c] + bias[c]
// R = 256, K = 384, C = 384.  One wave32 per 16x16 output tile, fp32 WMMA
// (V_WMMA_F32_16X16X4_F32), 96 K-steps of 4.  EXEC is all-ones (blockDim=32).
// ---------------------------------------------------------------------------
__global__ __launch_bounds__(32)
void wmma_gemm_f32(const float* __restrict__ A, const float* __restrict__ W,
                   const float* __restrict__ bias, float* __restrict__ out)
{
    const int lane = threadIdx.x;            // 0..31
    const int col0 = blockIdx.x * 16;
    const int row0 = blockIdx.y * 16;
    const int mn   = lane & 15;              // A row within tile == B/D column
    const int half = lane >> 4;              // K sub-pair selector

    v8f c = {};
    const float* __restrict__ arow = A + (size_t)(row0 + mn) * kD;
    for (int k0 = 0; k0 < kD; k0 += 4) {
        const int kb = k0 + half * 2;
        // A 16x4 layout: lanes 0-15 hold K={k0,k0+1}, lanes 16-31 K={k0+2,k0+3}
        v2f a;
        a.x = arow[kb];
        a.y = arow[kb + 1];
        // B 4x16 layout (mirror): lane column = mn, same K sub-pair
        v2f b;
        b.x = W[(size_t)(kb)     * kD + col0 + mn];
        b.y = W[(size_t)(kb + 1) * kD + col0 + mn];
        c = __builtin_amdgcn_wmma_f32_16x16x4_f32(
                /*neg_a=*/false, a, /*neg_b=*/false, b,
                /*c_mod=*/(short)0, c, /*reuse_a=*/false, /*reuse_b=*/false);
    }

    const float bv = bias[col0 + mn];
    #pragma unroll
    for (int r = 0; r < 8; ++r) {
        const int m = r + 8 * half;          // C/D layout: lanes16-31 hold M=8..15
        out[(size_t)(row0 + m) * kD + col0 + mn] = c[r] + bv;
    }
}

// ---------------------------------------------------------------------------
// Per-(b,t): scores GEMV, softmax, top-16 (lowest-index tie-break),
// renormalize, weighted sum of the 16 selected patch rows.
// One 256-thread block (8 wave32s) per (b,t).
// ---------------------------------------------------------------------------
__global__ __launch_bounds__(256)
void attn_topk_kernel(const float* __restrict__ qproj,  // [256, 384]
                      const float* __restrict__ pf,     // [256, 256, 384]
                      float* __restrict__ outv)         // [256, 384]
{
    __shared__ float s_q[kD];
    __shared__ float s_p[kNP];         // scores -> softmax probs (mutated by topk)
    __shared__ float s_rv[256];
    __shared__ int   s_ri[256];
    __shared__ float s_selw[kTopK];
    __shared__ int   s_seli[kTopK];
    __shared__ float s_inv;

    const int row = blockIdx.x;        // flattened (b,t)
    const int tid = threadIdx.x;
    const float* __restrict__ P = pf + (size_t)row * kNP * kD;

    // stage q_proj row in LDS
    for (int i = tid; i < kD; i += 256) s_q[i] = qproj[(size_t)row * kD + i];
    __syncthreads();

    // ---- scores[n] = dot(q, P[n,:])  (TEMP == 1) ----
    const int wave = tid >> 5, lane = tid & 31;
    for (int n = wave; n < kNP; n += 8) {
        const float* __restrict__ pr = P + (size_t)n * kD;
        if (n + 8 < kNP) __builtin_prefetch(P + (size_t)(n + 8) * kD + lane * 4, 0, 0);
        float acc = 0.f;
        for (int d = lane; d < kD; d += 32) acc += s_q[d] * pr[d];
        #pragma unroll
        for (int off = 16; off > 0; off >>= 1) acc += __shfl_down(acc, off, 32);
        if (lane == 0) s_p[n] = acc;
    }
    __syncthreads();

    // ---- softmax over 256 scores: thread tid owns score tid ----
    const float v = s_p[tid];
    s_rv[tid] = v; __syncthreads();
    for (int s = 128; s > 0; s >>= 1) {
        if (tid < s) s_rv[tid] = fmaxf(s_rv[tid], s_rv[tid + s]);
        __syncthreads();
    }
    const float mx = s_rv[0]; __syncthreads();
    const float e = expf(v - mx);
    s_rv[tid] = e; __syncthreads();
    for (int s = 128; s > 0; s >>= 1) {
        if (tid < s) s_rv[tid] += s_rv[tid + s];
        __syncthreads();
    }
    const float denom = s_rv[0]; __syncthreads();
    s_p[tid] = e / denom;
    __syncthreads();

    // ---- top-16 by iterative block argmax (ties -> lowest index) ----
    for (int it = 0; it < kTopK; ++it) {
        s_rv[tid] = s_p[tid]; s_ri[tid] = tid;
        __syncthreads();
        for (int s = 128; s > 0; s >>= 1) {
            if (tid < s) {
                const float vo = s_rv[tid + s]; const int io = s_ri[tid + s];
                if (vo > s_rv[tid] || (vo == s_rv[tid] && io < s_ri[tid])) {
                    s_rv[tid] = vo; s_ri[tid] = io;
                }
            }
            __syncthreads();
        }
        if (tid == 0) {
            const int bi = s_ri[0];
            s_seli[it] = bi;
            s_selw[it] = s_rv[0];
            s_p[bi] = -1.0f;               // remove from further consideration
        }
        __syncthreads();
    }
    if (tid == 0) {
        float ssum = 0.f;
        #pragma unroll
        for (int i = 0; i < kTopK; ++i) ssum += s_selw[i];
        s_inv = 1.0f / (ssum + 1e-8f);
    }
    __syncthreads();
    const float inv = s_inv;

    // ---- weighted sum of the 16 selected patch rows ----
    for (int d = tid; d < kD; d += 256) {
        float acc = 0.f;
        #pragma unroll
        for (int i = 0; i < kTopK; ++i)
            acc += s_selw[i] * P[(size_t)s_seli[i] * kD + d];
        outv[(size_t)row * kD + d] = acc * inv;
    }
}

// ---------------------------------------------------------------------------
extern "C" void kernel_launch(void* const* d_in, const int* in_sizes, int n_in,
                              void* d_out, int out_size, void* d_ws, size_t ws_size,
                              hipStream_t stream)
{
    const float* queries = (const float*)d_in[0];   // [B,T,QDIM]
    const float* pf      = (const float*)d_in[1];   // [B,T,N,D]
    const float* W_in    = (const float*)d_in[2];   // [QDIM,D]
    const float* b_in    = (const float*)d_in[3];   // [D]
    const float* W_out   = (const float*)d_in[4];   // [D,QDIM]
    const float* b_out   = (const float*)d_in[5];   // [QDIM]
    float* out = (float*)d_out;                     // [B,T,QDIM]

    float* qproj = (float*)d_ws;                    // 256*384 f32
    float* attnv = qproj + (size_t)kBT * kD;        // 256*384 f32

    dim3 gemm_grid(kD / 16, kBT / 16);              // 24 x 16 tiles
    wmma_gemm_f32<<<gemm_grid, 32, 0, stream>>>(queries, W_in, b_in, qproj);
    attn_topk_kernel<<<kBT, 256, 0, stream>>>(qproj, pf, attnv);
    wmma_gemm_f32<<<gemm_grid, 32, 0, stream>>>(attnv, W_out, b_out, out);
}